// GCN_8787503087873
// MI455X (gfx1250) — compile-verified
//
#include <hip/hip_runtime.h>

typedef __attribute__((ext_vector_type(2))) float v2f;
typedef __attribute__((ext_vector_type(8))) float v8f;

#define N_NODES 100000
#define N_EDGES 3200000
#define IN_F 512
#define HID 16
#define N_CLS 7

// ---------------- init kernels ----------------
__global__ void k_init_deg(float* __restrict__ deg) {
    int i = blockIdx.x * blockDim.x + threadIdx.x;
    if (i < N_NODES) deg[i] = 1.0f;  // self-loop contributes 1 to every degree
}

__global__ void k_init_agg(float* __restrict__ agg) {
    int i = blockIdx.x * blockDim.x + threadIdx.x;
    if (i < N_NODES * HID) agg[i] = 0.0f;
}

__global__ void k_init_out(const float* __restrict__ b2, float* __restrict__ out) {
    int i = blockIdx.x * blockDim.x + threadIdx.x;
    if (i < N_NODES * N_CLS) out[i] = b2[i % N_CLS];  // out = segment_sum + b2
}

__global__ void k_init_w2pad(const float* __restrict__ W2, float* __restrict__ W2p) {
    int i = threadIdx.x;  // 256 threads, one block
    int k = i >> 4, n = i & 15;
    W2p[i] = (n < N_CLS) ? W2[k * N_CLS + n] : 0.0f;
}

// ---------------- degree / norm ----------------
__global__ void k_degree(const long long* __restrict__ ei, float* __restrict__ deg) {
    int e = blockIdx.x * blockDim.x + threadIdx.x;
    if (e < N_EDGES) {
        int c = (int)ei[(size_t)N_EDGES + e];  // col = edge_index[1][e]
        atomicAdd(&deg[c], 1.0f);
    }
}

__global__ void k_rsqrt(float* __restrict__ deg) {
    int i = blockIdx.x * blockDim.x + threadIdx.x;
    if (i < N_NODES) deg[i] = rsqrtf(deg[i]);  // deg >= 1 always (self loops)
}

// ---------------- GEMM1: H1 = X @ W1 (WMMA f32 16x16x4) ----------------
// One wave32 per 16-node tile; W1 staged in LDS, shared by 8 waves/block.
__global__ void k_gemm1(const float* __restrict__ X, const float* __restrict__ W1,
                        float* __restrict__ H1) {
    __shared__ float Wl[IN_F * HID];  // 32 KB
    for (int i = threadIdx.x; i < IN_F * HID; i += blockDim.x) Wl[i] = W1[i];
    __syncthreads();

    int lane = threadIdx.x & 31;
    int half = lane >> 4;      // 0: K pair {0,1}, 1: K pair {2,3}
    int l    = lane & 15;      // A: row index; B/D: column index
    int tile = (blockIdx.x * blockDim.x + threadIdx.x) >> 5;
    int row0 = tile * 16;
    if (row0 >= N_NODES) return;  // wave-uniform: EXEC stays all-1s for WMMA

    const float* xrow = X + (size_t)(row0 + l) * IN_F;
    v8f c = {0.f, 0.f, 0.f, 0.f, 0.f, 0.f, 0.f, 0.f};

#pragma unroll 4
    for (int k = 0; k < IN_F; k += 4) {
        int kk = k + 2 * half;
        float2 af = *(const float2*)(xrow + kk);  // A: X[row0+l][kk], [kk+1]
        v2f a; a.x = af.x; a.y = af.y;
        v2f b; b.x = Wl[kk * HID + l]; b.y = Wl[(kk + 1) * HID + l];
        c = __builtin_amdgcn_wmma_f32_16x16x4_f32(false, a, false, b,
                                                  (short)0, c, false, false);
    }
    // D layout: VGPR r -> M = r + 8*half, N = l
    float* hrow = H1 + (size_t)row0 * HID;
#pragma unroll
    for (int r = 0; r < 8; ++r)
        hrow[(size_t)(r + 8 * half) * HID + l] = c[r];
}

// ---------------- scatter layer 1: agg[col] += H1[row] * norm ----------------
__global__ void k_scatter1(const long long* __restrict__ ei, const float* __restrict__ dis,
                           const float* __restrict__ H1, float* __restrict__ agg) {
    int e = blockIdx.x * blockDim.x + threadIdx.x;
    if (e >= N_EDGES + N_NODES) return;
    int r, cdst;
    if (e < N_EDGES) {
        r    = (int)ei[e];
        cdst = (int)ei[(size_t)N_EDGES + e];
    } else {
        r = cdst = e - N_EDGES;  // implicit self loop
    }
    float nrm = dis[r] * dis[cdst];
    const float4* hp = (const float4*)(H1 + (size_t)r * HID);
    float* ap = agg + (size_t)cdst * HID;
#pragma unroll
    for (int q = 0; q < 4; ++q) {
        float4 v = hp[q];
        atomicAdd(ap + 4 * q + 0, v.x * nrm);
        atomicAdd(ap + 4 * q + 1, v.y * nrm);
        atomicAdd(ap + 4 * q + 2, v.z * nrm);
        atomicAdd(ap + 4 * q + 3, v.w * nrm);
    }
}

// ---------------- GEMM2: H2 = relu(agg + b1) @ W2pad (WMMA) ----------------
// H2 stored padded to 8 floats/node for float4 loads in scatter2.
__global__ void k_gemm2(const float* __restrict__ agg, const float* __restrict__ b1,
                        const float* __restrict__ W2p, float* __restrict__ H2) {
    int lane = threadIdx.x & 31;
    int half = lane >> 4;
    int l    = lane & 15;
    int tile = (blockIdx.x * blockDim.x + threadIdx.x) >> 5;
    int row0 = tile * 16;
    if (row0 >= N_NODES) return;

    const float* arow = agg + (size_t)(row0 + l) * HID;
    v8f c = {0.f, 0.f, 0.f, 0.f, 0.f, 0.f, 0.f, 0.f};
#pragma unroll
    for (int k = 0; k < HID; k += 4) {
        int kk = k + 2 * half;
        v2f a;
        a.x = fmaxf(arow[kk] + b1[kk], 0.0f);          // fused bias + relu
        a.y = fmaxf(arow[kk + 1] + b1[kk + 1], 0.0f);
        v2f b;
        b.x = W2p[kk * 16 + l];
        b.y = W2p[(kk + 1) * 16 + l];
        c = __builtin_amdgcn_wmma_f32_16x16x4_f32(false, a, false, b,
                                                  (short)0, c, false, false);
    }
    if (l < N_CLS) {  // divergence only after all WMMA issued
        float* hrow = H2 + (size_t)row0 * 8;
#pragma unroll
        for (int r = 0; r < 8; ++r)
            hrow[(size_t)(r + 8 * half) * 8 + l] = c[r];
    }
}

// ---------------- scatter layer 2: out[col] += H2[row] * norm ----------------
__global__ void k_scatter2(const long long* __restrict__ ei, const float* __restrict__ dis,
                           const float* __restrict__ H2, float* __restrict__ out) {
    int e = blockIdx.x * blockDim.x + threadIdx.x;
    if (e >= N_EDGES + N_NODES) return;
    int r, cdst;
    if (e < N_EDGES) {
        r    = (int)ei[e];
        cdst = (int)ei[(size_t)N_EDGES + e];
    } else {
        r = cdst = e - N_EDGES;
    }
    float nrm = dis[r] * dis[cdst];
    const float4* hp = (const float4*)(H2 + (size_t)r * 8);
    float4 v0 = hp[0];
    float4 v1 = hp[1];  // .w is padding, never used
    float* op = out + (size_t)cdst * N_CLS;
    atomicAdd(op + 0, v0.x * nrm);
    atomicAdd(op + 1, v0.y * nrm);
    atomicAdd(op + 2, v0.z * nrm);
    atomicAdd(op + 3, v0.w * nrm);
    atomicAdd(op + 4, v1.x * nrm);
    atomicAdd(op + 5, v1.y * nrm);
    atomicAdd(op + 6, v1.z * nrm);
}

extern "C" void kernel_launch(void* const* d_in, const int* in_sizes, int n_in,
                              void* d_out, int out_size, void* d_ws, size_t ws_size,
                              hipStream_t stream) {
    const float*     x  = (const float*)d_in[0];
    const long long* ei = (const long long*)d_in[1];  // int64 edge_index [2, E]
    const float*     W1 = (const float*)d_in[2];
    const float*     b1 = (const float*)d_in[3];
    const float*     W2 = (const float*)d_in[4];
    const float*     b2 = (const float*)d_in[5];
    float* out = (float*)d_out;

    // workspace layout (floats); all offsets are 16B-aligned
    float* ws   = (float*)d_ws;
    float* dis  = ws;                              // N       (deg -> rsqrt in place)
    float* H1   = dis + N_NODES;                   // N*16
    float* agg  = H1 + (size_t)N_NODES * HID;      // N*16
    float* H2   = agg + (size_t)N_NODES * HID;     // N*8 (7 valid + 1 pad)
    float* W2p  = H2 + (size_t)N_NODES * 8;        // 256

    const int T = 256;
    int gN    = (N_NODES + T - 1) / T;
    int gNH   = (N_NODES * HID + T - 1) / T;
    int gNC   = (N_NODES * N_CLS + T - 1) / T;
    int gE    = (N_EDGES + T - 1) / T;
    int gEN   = (N_EDGES + N_NODES + T - 1) / T;
    int tiles = (N_NODES + 15) / 16;
    int gG    = (tiles + (T / 32) - 1) / (T / 32);  // 8 wave-tiles per block

    k_init_deg  <<<gN,  T, 0, stream>>>(dis);
    k_init_agg  <<<gNH, T, 0, stream>>>(agg);
    k_init_out  <<<gNC, T, 0, stream>>>(b2, out);
    k_init_w2pad<<<1,   T, 0, stream>>>(W2, W2p);

    k_degree    <<<gE,  T, 0, stream>>>(ei, dis);
    k_rsqrt     <<<gN,  T, 0, stream>>>(dis);

    k_gemm1     <<<gG,  T, 0, stream>>>(x, W1, H1);
    k_scatter1  <<<gEN, T, 0, stream>>>(ei, dis, H1, agg);
    k_gemm2     <<<gG,  T, 0, stream>>>(agg, b1, W2p, H2);
    k_scatter2  <<<gEN, T, 0, stream>>>(ei, dis, H2, out);
}